// DilatedKNN_1468878815323
// MI455X (gfx1250) — compile-verified
//
#include <hip/hip_runtime.h>

typedef __attribute__((ext_vector_type(2))) float v2f;
typedef __attribute__((ext_vector_type(8))) float v8f;

#define N_PTS   8192
#define CDIM    64
#define MTILE   64      // query rows per workgroup
#define NCHUNK  64      // support rows per chunk
#define KK      18      // k * dilation
#define KOUT    9       // k
#define QSTRIDE 68      // LDS row stride for Q tiles (16B-aligned float4, conflict-free frags)
#define DSTRIDE 65      // LDS row stride for dot tile (conflict-free per-row scan)

__global__ __launch_bounds__(64)
void dilated_knn_kernel(const float* __restrict__ q, int* __restrict__ out, int npts) {
  __shared__ float sQm[MTILE * QSTRIDE];
  __shared__ float sQn[NCHUNK * QSTRIDE];
  __shared__ float sD [MTILE * DSTRIDE];
  __shared__ float sSqN[NCHUNK];

  const int t = threadIdx.x;          // 0..63
  const int w = t >> 5;               // wave id 0..1
  const int l = t & 31;               // lane id
  const float* qb = q + (size_t)blockIdx.y * npts * CDIM;
  const int mbase = blockIdx.x * MTILE;

  // ---- load resident Qm tile (coalesced float4) ----
  {
    const float4* src = (const float4*)(qb + (size_t)mbase * CDIM);
    #pragma unroll
    for (int it = 0; it < (MTILE * CDIM / 4) / 64; ++it) {
      int idx = it * 64 + t;
      float4 v = src[idx];
      *(float4*)&sQm[(idx >> 4) * QSTRIDE + (idx & 15) * 4] = v;
    }
  }
  __syncthreads();

  // ---- |q_m|^2 for the row this thread owns ----
  float sqm = 0.f;
  #pragma unroll
  for (int c = 0; c < CDIM; c += 4) {
    float4 v = *(const float4*)&sQm[t * QSTRIDE + c];
    sqm = fmaf(v.x, v.x, sqm); sqm = fmaf(v.y, v.y, sqm);
    sqm = fmaf(v.z, v.z, sqm); sqm = fmaf(v.w, v.w, sqm);
  }

  // ---- hoist A fragments: wave w covers M-subtiles {2w, 2w+1} (rows 32w..32w+31)
  // A 16x4 f32 layout: lanes 0-15 hold (K=k0,k0+1), lanes 16-31 hold (K=k0+2,k0+3), M=lane&15
  v2f a0[16], a1[16];
  {
    const int arow = w * 32 + (l & 15);
    const int acol = ((l >> 4) & 1) * 2;
    #pragma unroll
    for (int k = 0; k < 16; ++k) {
      a0[k] = *(const v2f*)&sQm[arow        * QSTRIDE + k * 4 + acol];
      a1[k] = *(const v2f*)&sQm[(arow + 16) * QSTRIDE + k * 4 + acol];
    }
  }

  float best_d[KK];
  int   best_i[KK];
  #pragma unroll
  for (int i = 0; i < KK; ++i) { best_d[i] = 3.4e38f; best_i[i] = 0x7fffffff; }

  for (int chunk = 0; chunk < N_PTS / NCHUNK; ++chunk) {
    __syncthreads();  // previous consumers of sD/sSqN/sQn are done

    // ---- load Qn chunk (coalesced float4) + prefetch next chunk ----
    {
      const float4* src = (const float4*)(qb + (size_t)chunk * NCHUNK * CDIM);
      #pragma unroll
      for (int it = 0; it < (NCHUNK * CDIM / 4) / 64; ++it) {
        int idx = it * 64 + t;
        float4 v = src[idx];
        *(float4*)&sQn[(idx >> 4) * QSTRIDE + (idx & 15) * 4] = v;
      }
      if (chunk + 1 < N_PTS / NCHUNK) {
        const float* nxt = qb + (size_t)(chunk + 1) * NCHUNK * CDIM + t * 64;
        __builtin_prefetch(nxt, 0, 3);          // global_prefetch_b8
        __builtin_prefetch(nxt + 32, 0, 3);
      }
    }
    __syncthreads();

    // ---- |q_n|^2 for this chunk ----
    {
      float s = 0.f;
      #pragma unroll
      for (int c = 0; c < CDIM; c += 4) {
        float4 v = *(const float4*)&sQn[t * QSTRIDE + c];
        s = fmaf(v.x, v.x, s); s = fmaf(v.y, v.y, s);
        s = fmaf(v.z, v.z, s); s = fmaf(v.w, v.w, s);
      }
      sSqN[t] = s;
    }

    // ---- WMMA: dot(m, n) tiles -> sD ----
    {
      const int mrow = 32 * w + ((l >> 4) & 1) * 8;  // C-layout: lanes>=16 -> M += 8
      const int ncl  = l & 15;
      const int bcol = ((l >> 4) & 1) * 2;           // B 4x16 layout mirrors A
      #pragma unroll
      for (int nj = 0; nj < NCHUNK / 16; ++nj) {
        v8f acc0 = {};
        v8f acc1 = {};
        const int bbase = (nj * 16 + ncl) * QSTRIDE + bcol;
        #pragma unroll
        for (int k = 0; k < 16; ++k) {
          v2f bf = *(const v2f*)&sQn[bbase + k * 4];
          acc0 = __builtin_amdgcn_wmma_f32_16x16x4_f32(
              false, a0[k], false, bf, (short)0, acc0, false, false);
          acc1 = __builtin_amdgcn_wmma_f32_16x16x4_f32(
              false, a1[k], false, bf, (short)0, acc1, false, false);
        }
        const int ncol = nj * 16 + ncl;
        #pragma unroll
        for (int r = 0; r < 8; ++r) {
          sD[(mrow + r)      * DSTRIDE + ncol] = acc0[r];
          sD[(mrow + 16 + r) * DSTRIDE + ncol] = acc1[r];
        }
      }
    }
    __syncthreads();

    // ---- top-18 selection: thread t owns local row t ----
    const int gbase = chunk * NCHUNK;
    for (int n = 0; n < NCHUNK; ++n) {
      float dot = sD[t * DSTRIDE + n];
      float d2  = fmaf(-2.f, dot, sqm + sSqN[n]);
      d2 = fmaxf(d2, 0.f);                 // match reference's max(d2,0) tie behavior
      if (d2 < best_d[KK - 1]) {           // rare path: branchless sorted insert
        float cd = d2;
        int   ci = gbase + n;
        #pragma unroll
        for (int j = 0; j < KK; ++j) {
          bool lt  = cd < best_d[j];       // strict < => stable ties (lowest index first)
          float td = best_d[j]; int ti = best_i[j];
          best_d[j] = lt ? cd : td;  best_i[j] = lt ? ci : ti;
          cd        = lt ? td : cd;  ci        = lt ? ti : ci;
        }
      }
    }
  }

  // ---- dilated slice ::2 of the 18-NN list ----
  int* o = out + ((size_t)blockIdx.y * npts + (mbase + t)) * KOUT;
  #pragma unroll
  for (int k = 0; k < KOUT; ++k) o[k] = best_i[2 * k];
}

extern "C" void kernel_launch(void* const* d_in, const int* in_sizes, int n_in,
                              void* d_out, int out_size, void* d_ws, size_t ws_size,
                              hipStream_t stream) {
  (void)n_in; (void)d_ws; (void)ws_size; (void)out_size;
  const float* q = (const float*)d_in[0];
  int* out = (int*)d_out;
  const int batches = in_sizes[0] / (N_PTS * CDIM);   // 4
  dim3 grid(N_PTS / MTILE, batches);
  dilated_knn_kernel<<<grid, 64, 0, stream>>>(q, out, N_PTS);
}